// LocalSpatialEncoding_81870666596685
// MI455X (gfx1250) — compile-verified
//
#include <hip/hip_runtime.h>

typedef float v2f __attribute__((ext_vector_type(2)));
typedef float v8f __attribute__((ext_vector_type(8)));

#define BB    2
#define NN    65536
#define KK    16
#define NMOM  35            // 7 first moments + 28 upper-tri second moments
#define MOMPAD 40
#define K1_BLOCKS 1024
#define FOLD_OFF  65536     // float offset into workspace for folded params
#define BN_EPS 1e-6f

// ---------------------------------------------------------------------------
// Kernel 1: per-block partial moments of v = (cx,cy,cz,nx,ny,nz,dist)
// ---------------------------------------------------------------------------
__global__ void lse_k1_moments(const float* __restrict__ xyz,
                               const int*   __restrict__ idx,
                               float*       __restrict__ ws) {
  float s[NMOM];
#pragma unroll
  for (int i = 0; i < NMOM; ++i) s[i] = 0.f;

  const int total  = BB * NN * KK;
  const int stride = gridDim.x * blockDim.x;
  for (int smp = blockIdx.x * blockDim.x + threadIdx.x; smp < total; smp += stride) {
    int b   = smp >> 20;                 // N*K = 2^20
    int rem = smp & ((1 << 20) - 1);
    int n   = rem >> 4;                  // K = 16
    int j   = idx[smp];                  // flat (B,N,K) == smp order
    const float* xb = xyz + b * 3 * NN;
    float cx = xb[n],      cy = xb[NN + n],  cz = xb[2 * NN + n];
    float nx = xb[j],      ny = xb[NN + j],  nz = xb[2 * NN + j];
    float dx = cx - nx, dy = cy - ny, dz = cz - nz;
    float dist = sqrtf(dx * dx + dy * dy + dz * dz);
    float v[7] = {cx, cy, cz, nx, ny, nz, dist};
    int c = 7;
#pragma unroll
    for (int i = 0; i < 7; ++i) {
      s[i] += v[i];
#pragma unroll
      for (int jj = i; jj < 7; ++jj) s[c++] += v[i] * v[jj];
    }
  }

  // wave32 butterfly reduce
#pragma unroll
  for (int i = 0; i < NMOM; ++i) {
    float x = s[i];
    for (int m = 16; m >= 1; m >>= 1) x += __shfl_xor(x, m, 32);
    s[i] = x;
  }

  __shared__ float lds[8 * NMOM];
  const int lane = threadIdx.x & 31;
  const int wv   = threadIdx.x >> 5;
  if (lane == 0) {
#pragma unroll
    for (int i = 0; i < NMOM; ++i) lds[wv * NMOM + i] = s[i];
  }
  __syncthreads();
  if (threadIdx.x < NMOM) {
    float acc = 0.f;
    for (int w = 0; w < 8; ++w) acc += lds[w * NMOM + threadIdx.x];
    ws[blockIdx.x * MOMPAD + threadIdx.x] = acc;   // deterministic partial
  }
}

// ---------------------------------------------------------------------------
// Kernel 2: fixed-order final reduce + fold BN into weights/bias
// ---------------------------------------------------------------------------
__global__ void lse_k2_finalize(const float* __restrict__ W,
                                const float* __restrict__ gamma,
                                const float* __restrict__ beta,
                                float*       __restrict__ ws) {
  __shared__ float S[NMOM];
  const int t = threadIdx.x;
  if (t < NMOM) {
    float a = 0.f;
    for (int p = 0; p < K1_BLOCKS; ++p) a += ws[p * MOMPAD + t];
    S[t] = a;
  }
  __syncthreads();
  if (t < 32) {
    float w[7];
    w[0] = W[t * 10 + 0] + W[t * 10 + 6];
    w[1] = W[t * 10 + 1] + W[t * 10 + 7];
    w[2] = W[t * 10 + 2] + W[t * 10 + 8];
    w[3] = W[t * 10 + 3] - W[t * 10 + 6];
    w[4] = W[t * 10 + 4] - W[t * 10 + 7];
    w[5] = W[t * 10 + 5] - W[t * 10 + 8];
    w[6] = W[t * 10 + 9];
    const float invM = 1.f / (float)(BB * NN * KK);
    float mean = 0.f;
#pragma unroll
    for (int i = 0; i < 7; ++i) mean += w[i] * S[i];
    mean *= invM;
    float eh2 = 0.f;
    int c = 7;
#pragma unroll
    for (int i = 0; i < 7; ++i)
#pragma unroll
      for (int jj = i; jj < 7; ++jj) {
        float coef = (i == jj) ? w[i] * w[i] : 2.f * w[i] * w[jj];
        eh2 += coef * S[c++];
      }
    eh2 *= invM;
    float var = eh2 - mean * mean;
    float sc  = gamma[t] * rsqrtf(var + BN_EPS);
    float bi  = beta[t] - sc * mean;
    float* F = ws + FOLD_OFF;
    F[t * 4 + 0] = sc * w[0];           // A-matrix 1: (cx,cy,cz,dist)
    F[t * 4 + 1] = sc * w[1];
    F[t * 4 + 2] = sc * w[2];
    F[t * 4 + 3] = sc * w[6];
    F[128 + t * 4 + 0] = sc * w[3];     // A-matrix 2: (nx,ny,nz,0)
    F[128 + t * 4 + 1] = sc * w[4];
    F[128 + t * 4 + 2] = sc * w[5];
    F[128 + t * 4 + 3] = 0.f;
    F[256 + t] = bi;                    // folded bias
  }
}

// ---------------------------------------------------------------------------
// Kernel 3: WMMA pass — per point, 16 neighbors = 16 columns, 2 channel tiles
// ---------------------------------------------------------------------------
__global__ void lse_k3_main(const float* __restrict__ xyz,
                            const float* __restrict__ feat,
                            const int*   __restrict__ idx,
                            const float* __restrict__ ws,
                            float*       __restrict__ out) {
  const int  lane = threadIdx.x & 31;
  const bool hi   = lane >= 16;
  const int  l15  = lane & 15;
  const float* F  = ws + FOLD_OFF;
  const float inf = __builtin_inff();

  // A-matrices (16x4 f32): VGPR0 = K0|K2 split at lane16, VGPR1 = K1|K3
  v2f a1[2], a2[2];
  v8f cb[2];
#pragma unroll
  for (int tt = 0; tt < 2; ++tt) {
    int ch = tt * 16 + l15;
    a1[tt].x = hi ? F[ch * 4 + 2]       : F[ch * 4 + 0];
    a1[tt].y = hi ? F[ch * 4 + 3]       : F[ch * 4 + 1];
    a2[tt].x = hi ? F[128 + ch * 4 + 2] : F[128 + ch * 4 + 0];
    a2[tt].y = hi ? F[128 + ch * 4 + 3] : F[128 + ch * 4 + 1];
#pragma unroll
    for (int r = 0; r < 8; ++r)          // C layout: VGPR r = rows r | r+8
      cb[tt][r] = F[256 + tt * 16 + r + (hi ? 8 : 0)];
  }

  const int nwaves = (gridDim.x * blockDim.x) >> 5;
  const int wid    = (blockIdx.x * blockDim.x + threadIdx.x) >> 5;

  for (int task = wid; task < BB * NN; task += nwaves) {   // task = b*N + n
    int b = task >> 16;
    int n = task & (NN - 1);
    const float* xb = xyz + b * 3 * NN;
    int j = idx[(task << 4) + l15];
    float cx = xb[n], cy = xb[NN + n], cz = xb[2 * NN + n];
    float nx = xb[j], ny = xb[NN + j], nz = xb[2 * NN + j];
    float dx = cx - nx, dy = cy - ny, dz = cz - nz;
    float dist = sqrtf(dx * dx + dy * dy + dz * dz);

    // B-matrices (4x16 f32): VGPR0 = K0|K2 split at lane16, VGPR1 = K1|K3
    v2f b1, b2;
    b1.x = hi ? cz   : cx;
    b1.y = hi ? dist : cy;
    b2.x = hi ? nz   : nx;
    b2.y = hi ? 0.f  : ny;

    // per-task base element index: (b<<26) + (n<<4) + l15 (+ row-half offset)
    const int base = (b << 26) + (n << 4) + l15 + (hi ? (8 << 20) : 0);

#pragma unroll
    for (int tt = 0; tt < 2; ++tt) {
      v8f c = cb[tt];
      c = __builtin_amdgcn_wmma_f32_16x16x4_f32(false, a1[tt], false, b1,
                                                (short)0, c, false, false);
      c = __builtin_amdgcn_wmma_f32_16x16x4_f32(false, a2[tt], false, b2,
                                                (short)0, c, false, false);
#pragma unroll
      for (int r = 0; r < 8; ++r) {
        float v = __builtin_amdgcn_fmed3f(c[r], 0.f, inf);  // single v_med3 ReLU
        out[base + ((tt * 16 + r) << 20)] = v;
      }
    }

    // feat passthrough: out channels 32..63, value uniform over k -> b128 stores
    {
      int chq = lane >> 2;               // 0..7
      int kb  = (lane & 3) << 2;         // 0,4,8,12
      const int fbase = (b << 26) + (n << 4) + kb;
#pragma unroll
      for (int g = 0; g < 4; ++g) {
        int cch = g * 8 + chq;
        float v = feat[(b * 32 + cch) * NN + n];
        float4 vv = make_float4(v, v, v, v);
        *reinterpret_cast<float4*>(&out[fbase + ((32 + cch) << 20)]) = vv;
      }
    }
  }
}

// ---------------------------------------------------------------------------
extern "C" void kernel_launch(void* const* d_in, const int* in_sizes, int n_in,
                              void* d_out, int out_size, void* d_ws, size_t ws_size,
                              hipStream_t stream) {
  (void)in_sizes; (void)n_in; (void)out_size; (void)ws_size;
  const float* xyz   = (const float*)d_in[0];
  const float* feat  = (const float*)d_in[1];
  const int*   idx   = (const int*)  d_in[2];
  const float* W     = (const float*)d_in[3];
  const float* gamma = (const float*)d_in[4];
  const float* beta  = (const float*)d_in[5];
  float* out = (float*)d_out;
  float* ws  = (float*)d_ws;

  lse_k1_moments<<<K1_BLOCKS, 256, 0, stream>>>(xyz, idx, ws);
  lse_k2_finalize<<<1, 64, 0, stream>>>(W, gamma, beta, ws);
  lse_k3_main<<<2048, 256, 0, stream>>>(xyz, feat, idx, ws, out);
}